// RVQ_42829413876014
// MI455X (gfx1250) — compile-verified
//
#include <hip/hip_runtime.h>
#include <math.h>

// RVQ forward for MI455X (gfx1250, wave32, WMMA).
// x: [16,4096,256] f32, emb: [8,1024,256] f32
// out: quantized [16*4096*256] f32 | total_loss [1] f32 | indices [16*4096*8] f32

typedef float v2f __attribute__((ext_vector_type(2)));
typedef float v8f __attribute__((ext_vector_type(8)));

#define RVQ_N      65536      // B*T rows
#define RVQ_D      256
#define RVQ_K      1024
#define RVQ_Q      8
#define RVQ_WG     512        // main-kernel workgroups (128 rows each)

// ---------------------------------------------------------------- norms ----
__global__ __launch_bounds__(256) void rvq_norms_kernel(
    const float* __restrict__ emb, float* __restrict__ norms) {
  int k = blockIdx.x * blockDim.x + threadIdx.x;   // 0..8191 (Q*K)
  const float* e = emb + (size_t)k * RVQ_D;
  float s = 0.f;
  #pragma unroll 8
  for (int d = 0; d < RVQ_D; d += 4) {
    float4 v = *(const float4*)(e + d);
    s = fmaf(v.x, v.x, s); s = fmaf(v.y, v.y, s);
    s = fmaf(v.z, v.z, s); s = fmaf(v.w, v.w, s);
  }
  norms[k] = s;
}

// ----------------------------------------------------------------- main ----
__global__ __launch_bounds__(256) void rvq_main_kernel(
    const float* __restrict__ x, const float* __restrict__ emb,
    const float* __restrict__ norms, float* __restrict__ qout,
    float* __restrict__ idx_out, float* __restrict__ wg_loss) {
  __shared__ int   ldsIdx[8 * 16];   // per-wave chosen code per row
  __shared__ float ldsLoss[8];

  const int tid  = threadIdx.x;
  const int wave = tid >> 5;
  const int lane = tid & 31;
  const int l15  = lane & 15;
  const int hf   = lane >> 4;

  const int rowBase = blockIdx.x * 128 + wave * 16;
  const int myRow   = rowBase + l15;            // row this lane holds in A-frags

  // A-fragment layout (V_WMMA_F32_16X16X4_F32, 16x4 f32 A):
  //   lanes 0-15 hold M=lane, K={4j,4j+1}; lanes 16-31 hold M=lane-16, K={4j+2,4j+3}
  // -> lane owns row (lane&15), dims 4j + 2*(lane>>4) + {0,1} for fragment j.
  const float* xrow = x + (size_t)myRow * RVQ_D + 2 * hf;
  v2f a[64];
  #pragma unroll
  for (int j = 0; j < 64; ++j) a[j] = *(const v2f*)(xrow + 4 * j);

  float lossAcc = 0.f;

  for (int q = 0; q < RVQ_Q; ++q) {
    const float* eq = emb + (size_t)q * RVQ_K * RVQ_D;
    const float* nq = norms + q * RVQ_K;

    float best[8];
    int   bidx[8];
    #pragma unroll
    for (int s = 0; s < 8; ++s) { best[s] = INFINITY; bidx[s] = 0; }

    for (int c = 0; c < 64; ++c) {              // 64 chunks of 16 codes
      const int code = c * 16 + l15;
      // B-fragment: same per-lane float2 pattern, codes in place of rows.
      const float* erow = eq + (size_t)code * RVQ_D + 2 * hf;
      if (c + 1 < 64)                           // prefetch next chunk's codes
        __builtin_prefetch(erow + 16 * RVQ_D, 0, 0);

      v8f acc = {};
      #pragma unroll
      for (int jj = 0; jj < 4; ++jj) {
        v2f bf[16];
        #pragma unroll
        for (int t = 0; t < 16; ++t)
          bf[t] = *(const v2f*)(erow + 4 * (jj * 16 + t));
        #pragma unroll
        for (int t = 0; t < 16; ++t)
          acc = __builtin_amdgcn_wmma_f32_16x16x4_f32(
              false, a[jj * 16 + t], false, bf[t], (short)0, acc, false, false);
      }

      // score = ||e||^2 - 2 * (r . e)   (||r||^2 constant per row for argmin)
      const float nrm = nq[code];
      #pragma unroll
      for (int s = 0; s < 8; ++s) {
        float sc = fmaf(-2.f, acc[s], nrm);
        if (sc < best[s]) { best[s] = sc; bidx[s] = code; }
      }
    }

    // C-layout: vgpr s -> row s (lanes 0-15) / row 8+s (lanes 16-31).
    // Reduce argmin across each 16-lane half (xor offsets stay in-half).
    #pragma unroll
    for (int s = 0; s < 8; ++s) {
      float v  = best[s];
      int   bi = bidx[s];
      #pragma unroll
      for (int off = 8; off >= 1; off >>= 1) {
        float ov = __shfl_xor(v, off, 32);
        int   oi = __shfl_xor(bi, off, 32);
        if (ov < v || (ov == v && oi < bi)) { v = ov; bi = oi; }
      }
      if (lane == 0)  ldsIdx[wave * 16 + s]     = bi;  // row s
      if (lane == 16) ldsIdx[wave * 16 + 8 + s] = bi;  // row 8+s
    }
    __syncthreads();

    const int myIdx = ldsIdx[wave * 16 + l15];
    if (lane < 16)
      idx_out[(size_t)(rowBase + lane) * RVQ_Q + q] = (float)myIdx;

    // residual update r -= e[idx]; loss_i = 1.25 * mean(r_new^2)
    const float* crow = eq + (size_t)myIdx * RVQ_D + 2 * hf;
    float ls = 0.f;
    #pragma unroll
    for (int j = 0; j < 64; ++j) {
      v2f e = *(const v2f*)(crow + 4 * j);
      a[j].x -= e.x;
      a[j].y -= e.y;
      ls = fmaf(a[j].x, a[j].x, ls);
      ls = fmaf(a[j].y, a[j].y, ls);
    }
    lossAcc += ls;
    __syncthreads();
  }

  // quantized_out = x - r_final (telescoping sum of all zq)
  float* orow = qout + (size_t)myRow * RVQ_D + 2 * hf;
  #pragma unroll
  for (int j = 0; j < 64; ++j) {
    v2f xv = *(const v2f*)(xrow + 4 * j);
    v2f o; o.x = xv.x - a[j].x; o.y = xv.y - a[j].y;
    *(v2f*)(orow + 4 * j) = o;
  }

  // deterministic per-WG loss partial
  #pragma unroll
  for (int off = 16; off >= 1; off >>= 1)
    lossAcc += __shfl_xor(lossAcc, off, 32);
  if (lane == 0) ldsLoss[wave] = lossAcc;
  __syncthreads();
  if (tid == 0) {
    float s = 0.f;
    #pragma unroll
    for (int w = 0; w < 8; ++w) s += ldsLoss[w];
    wg_loss[blockIdx.x] = s;
  }
}

// ------------------------------------------------------------- finalize ----
__global__ void rvq_loss_finalize(const float* __restrict__ wg_loss,
                                  float* __restrict__ out_loss) {
  if (threadIdx.x == 0 && blockIdx.x == 0) {
    float s = 0.f;
    for (int i = 0; i < RVQ_WG; ++i) s += wg_loss[i];
    out_loss[0] = s * (1.25f / ((float)RVQ_N * (float)RVQ_D));
  }
}

// -------------------------------------------------------------- launch ----
extern "C" void kernel_launch(void* const* d_in, const int* in_sizes, int n_in,
                              void* d_out, int out_size, void* d_ws, size_t ws_size,
                              hipStream_t stream) {
  const float* x   = (const float*)d_in[0];   // [16,4096,256]
  const float* emb = (const float*)d_in[1];   // [8,1024,256]

  float* qout = (float*)d_out;                          // 16777216 floats
  float* loss = (float*)d_out + (size_t)RVQ_N * RVQ_D;  // 1 float
  float* idxo = loss + 1;                               // 524288 floats

  float* norms   = (float*)d_ws;          // Q*K = 8192 floats
  float* wg_loss = norms + RVQ_Q * RVQ_K; // RVQ_WG floats

  rvq_norms_kernel<<<(RVQ_Q * RVQ_K) / 256, 256, 0, stream>>>(emb, norms);
  rvq_main_kernel<<<RVQ_WG, 256, 0, stream>>>(x, emb, norms, qout, idxo, wg_loss);
  rvq_loss_finalize<<<1, 32, 0, stream>>>(wg_loss, loss);
}